// DAGGNN_29403346109071
// MI455X (gfx1250) — compile-verified
//
#include <hip/hip_runtime.h>
#include <hip/hip_bf16.h>
#include <math.h>

#define Bd   4
#define Nd   20000
#define NEVd 100000
#define Ed   400000
#define Dd   64
#define Ld   3

typedef __attribute__((ext_vector_type(2))) float v2f;
typedef __attribute__((ext_vector_type(8))) float v8f;

// ---------------- helpers ----------------

__device__ __forceinline__ void atomic_fadd(float* p, float v) {
    __hip_atomic_fetch_add(p, v, __ATOMIC_RELAXED, __HIP_MEMORY_SCOPE_AGENT);
}

__global__ void zero_f32(float* __restrict__ p, long n) {
    long i = (long)blockIdx.x * blockDim.x + threadIdx.x;
    long s = (long)gridDim.x * blockDim.x;
    for (; i < n; i += s) p[i] = 0.0f;
}

// ---------------- degree ----------------

__global__ void deg_kernel(float* __restrict__ deg, const int* __restrict__ dag) {
    int e = blockIdx.x * blockDim.x + threadIdx.x;
    if (e < Ed) atomic_fadd(&deg[dag[e]], 1.0f);   // dst = dag[0][e]
}

__global__ void rdeg_kernel(float* __restrict__ deg) {
    int e = blockIdx.x * blockDim.x + threadIdx.x;
    if (e < NEVd) deg[e] = 1.0f / fmaxf(deg[e], 1.0f);
}

// ---------------- event projection: h = relu(feat6 @ Wp + bp) ----------------

__global__ void proj_kernel(const float* __restrict__ x, const float* __restrict__ Wp,
                            const float* __restrict__ bp, const int* __restrict__ ev_to,
                            const int* __restrict__ ev_src, float* __restrict__ h) {
    long t = (long)blockIdx.x * blockDim.x + threadIdx.x;   // over B*NEV*64
    if (t >= (long)Bd * NEVd * Dd) return;
    int d = (int)(t & 63);
    long be = t >> 6;
    int e = (int)(be % NEVd);
    int b = (int)(be / NEVd);
    const float* xs = x + ((long)b * Nd + ev_src[e]) * 3;
    const float* xt = x + ((long)b * Nd + ev_to[e]) * 3;
    float acc = bp[d];
    acc += xs[0] * Wp[0 * Dd + d];
    acc += xs[1] * Wp[1 * Dd + d];
    acc += xs[2] * Wp[2 * Dd + d];
    acc += xt[0] * Wp[3 * Dd + d];
    acc += xt[1] * Wp[4 * Dd + d];
    acc += xt[2] * Wp[5 * Dd + d];
    h[t] = fmaxf(acc, 0.0f);
}

// ---------------- edge scatter: agg[:,dst] += h[:,src] ----------------

__global__ void scatter_kernel(float* __restrict__ agg, const float* __restrict__ h,
                               const int* __restrict__ dag) {
    long t = (long)blockIdx.x * blockDim.x + threadIdx.x;   // over E*64
    if (t >= (long)Ed * Dd) return;
    int d = (int)(t & 63);
    int e = (int)(t >> 6);
    int dst = dag[e];
    int src = dag[Ed + e];
#pragma unroll
    for (int b = 0; b < Bd; ++b) {
        float v = h[((long)b * NEVd + src) * Dd + d];
        atomic_fadd(&agg[((long)b * NEVd + dst) * Dd + d], v);
    }
}

// ---------------- fused layer GEMM via WMMA f32 16x16x4 ----------------
// hout = relu( hin @ Ws + (agg * rdeg_row) @ Wa + ba )
// 256 threads = 8 waves; each wave computes 16 rows x 64 cols.
// Weights staged transposed in LDS: lw[n*64 + k] = W[k*64 + n].

__global__ void __launch_bounds__(256)
gemm_layer_kernel(const float* __restrict__ hin, const float* __restrict__ agg,
                  const float* __restrict__ rdeg, const float* __restrict__ Ws,
                  const float* __restrict__ Wa, const float* __restrict__ ba,
                  float* __restrict__ hout) {
    __shared__ float lws[Dd * Dd];
    __shared__ float lwa[Dd * Dd];

    int tid = threadIdx.x;
    for (int idx = tid; idx < Dd * Dd; idx += 256) {
        int k = idx >> 6;
        int n = idx & 63;
        lws[n * Dd + k] = Ws[idx];
        lwa[n * Dd + k] = Wa[idx];
    }
    __syncthreads();

    int wave = tid >> 5;
    int lane = tid & 31;
    int mrow = lane & 15;        // M (for A) / N-in-tile (for B,C)
    int khalf = lane >> 4;       // which K pair this lane holds

    long rowBase = (long)blockIdx.x * 128 + wave * 16;   // flat row in [B*NEV)
    long r = rowBase + mrow;
    int e = (int)(r % NEVd);
    float scale = rdeg[e];

    const float* hrow = hin + r * Dd;
    const float* grow = agg + r * Dd;

    // A operands per ISA 16x4 f32 layout: lane m holds K = kk*4 + khalf*2 + {0,1}
    v2f ah[16], ag[16];
#pragma unroll
    for (int kk = 0; kk < 16; ++kk) {
        int koff = kk * 4 + khalf * 2;
        v2f a = *(const v2f*)(hrow + koff);
        v2f g = *(const v2f*)(grow + koff);
        g.x *= scale;
        g.y *= scale;
        ah[kk] = a;
        ag[kk] = g;
    }

#pragma unroll
    for (int nt = 0; nt < 4; ++nt) {
        int ncol = nt * 16 + mrow;
        float bias = ba[ncol];
        v8f acc = {};
#pragma unroll
        for (int kk = 0; kk < 16; ++kk) {
            int kb = kk * 4 + khalf * 2;
            v2f bs = *(const v2f*)(&lws[ncol * Dd + kb]);
            v2f bw = *(const v2f*)(&lwa[ncol * Dd + kb]);
            acc = __builtin_amdgcn_wmma_f32_16x16x4_f32(
                false, ah[kk], false, bs, (short)0, acc, false, false);
            acc = __builtin_amdgcn_wmma_f32_16x16x4_f32(
                false, ag[kk], false, bw, (short)0, acc, false, false);
        }
        // C/D layout: lane&15 -> N, VGPR v -> M = v + (lane>=16)*8
#pragma unroll
        for (int v = 0; v < 8; ++v) {
            int m = v + khalf * 8;
            hout[(rowBase + m) * Dd + nt * 16 + mrow] = fmaxf(acc[v] + bias, 0.0f);
        }
    }
}

// ---------------- node aggregation ----------------

__global__ void count_kernel(float* __restrict__ cnt, const int* __restrict__ ev_to) {
    int e = blockIdx.x * blockDim.x + threadIdx.x;
    if (e < NEVd) atomic_fadd(&cnt[ev_to[e]], 1.0f);
}

__global__ void nodeagg_kernel(float* __restrict__ nemb, const float* __restrict__ h,
                               const int* __restrict__ ev_to) {
    long t = (long)blockIdx.x * blockDim.x + threadIdx.x;   // over B*NEV*64
    if (t >= (long)Bd * NEVd * Dd) return;
    int d = (int)(t & 63);
    long be = t >> 6;
    int e = (int)(be % NEVd);
    int b = (int)(be / NEVd);
    int node = ev_to[e];
    atomic_fadd(&nemb[((long)b * Nd + node) * Dd + d], h[t]);
}

// ---------------- scores ----------------

__global__ void score_kernel(const float* __restrict__ nemb, const float* __restrict__ cnt,
                             const float* __restrict__ Wo, const float* __restrict__ bo,
                             const float* __restrict__ x, float* __restrict__ scores) {
    long t = (long)blockIdx.x * blockDim.x + threadIdx.x;   // over B*N
    if (t >= (long)Bd * Nd) return;
    int n = (int)(t % Nd);
    int b = (int)(t / Nd);
    float inv = 1.0f / fmaxf(cnt[n], 1.0f);
    const float* row = nemb + t * Dd;
    float acc = 0.0f;
#pragma unroll
    for (int d = 0; d < Dd; ++d) acc += row[d] * Wo[d];
    float s = acc * inv + bo[0];
    if (x[((long)b * Nd + n) * 3] > 0.0f) s = -INFINITY;
    scores[t] = s;
}

// ---------------- log_softmax per batch (block per b, wave32 reductions) ----------

__global__ void __launch_bounds__(1024)
logsoftmax_kernel(const float* __restrict__ scores, float* __restrict__ out) {
    __shared__ float red[32];
    __shared__ float bcast;
    int b = blockIdx.x;
    int tid = threadIdx.x;
    const float* s = scores + (long)b * Nd;

    // 1) max
    float m = -INFINITY;
    for (int i = tid; i < Nd; i += 1024) m = fmaxf(m, s[i]);
    for (int o = 16; o > 0; o >>= 1) m = fmaxf(m, __shfl_xor(m, o));
    if ((tid & 31) == 0) red[tid >> 5] = m;
    __syncthreads();
    if (tid < 32) {
        float v = red[tid];
        for (int o = 16; o > 0; o >>= 1) v = fmaxf(v, __shfl_xor(v, o));
        if (tid == 0) bcast = v;
    }
    __syncthreads();
    float mx = bcast;
    __syncthreads();

    // 2) sum of exp
    float acc = 0.0f;
    for (int i = tid; i < Nd; i += 1024) acc += expf(s[i] - mx);
    for (int o = 16; o > 0; o >>= 1) acc += __shfl_xor(acc, o);
    if ((tid & 31) == 0) red[tid >> 5] = acc;
    __syncthreads();
    if (tid < 32) {
        float v = red[tid];
        for (int o = 16; o > 0; o >>= 1) v += __shfl_xor(v, o);
        if (tid == 0) bcast = logf(v);
    }
    __syncthreads();
    float lse = bcast;

    // 3) write
    for (int i = tid; i < Nd; i += 1024) out[(long)b * Nd + i] = s[i] - mx - lse;
}

// ---------------- launcher ----------------

extern "C" void kernel_launch(void* const* d_in, const int* in_sizes, int n_in,
                              void* d_out, int out_size, void* d_ws, size_t ws_size,
                              hipStream_t stream) {
    const float* x   = (const float*)d_in[0];
    const float* Wp  = (const float*)d_in[1];
    const float* bp  = (const float*)d_in[2];
    const float* Ws  = (const float*)d_in[3];
    const float* Wa  = (const float*)d_in[4];
    const float* ba  = (const float*)d_in[5];
    const float* Wo  = (const float*)d_in[6];
    const float* bo  = (const float*)d_in[7];
    const int*   dag = (const int*)d_in[8];
    const int*   evt = (const int*)d_in[9];
    const int*   evs = (const int*)d_in[10];
    float* out = (float*)d_out;

    const long BNEVD = (long)Bd * NEVd * Dd;          // 25,600,000 floats
    char* wsb = (char*)d_ws;
    size_t off = 0;
    auto alloc = [&](size_t bytes) { void* p = wsb + off; off = (off + bytes + 255) & ~(size_t)255; return p; };
    float* h0   = (float*)alloc(BNEVD * 4);
    float* h1   = (float*)alloc(BNEVD * 4);
    float* agg  = (float*)alloc(BNEVD * 4);
    float* rdeg = (float*)alloc((size_t)NEVd * 4);
    float* nemb = (float*)alloc((size_t)Bd * Nd * Dd * 4);
    float* cnt  = (float*)alloc((size_t)Nd * 4);
    float* scr  = (float*)alloc((size_t)Bd * Nd * 4);

    // degrees
    zero_f32<<<4096, 256, 0, stream>>>(rdeg, NEVd);
    deg_kernel<<<(Ed + 255) / 256, 256, 0, stream>>>(rdeg, dag);
    rdeg_kernel<<<(NEVd + 255) / 256, 256, 0, stream>>>(rdeg);

    // projection
    proj_kernel<<<(int)((BNEVD + 255) / 256), 256, 0, stream>>>(x, Wp, bp, evt, evs, h0);

    // message-passing layers
    float* hin = h0;
    float* hout = h1;
    for (int l = 0; l < Ld; ++l) {
        zero_f32<<<4096, 256, 0, stream>>>(agg, BNEVD);
        scatter_kernel<<<(int)(((long)Ed * Dd + 255) / 256), 256, 0, stream>>>(agg, hin, dag);
        gemm_layer_kernel<<<(int)((long)Bd * NEVd / 128), 256, 0, stream>>>(
            hin, agg, rdeg, Ws + l * Dd * Dd, Wa + l * Dd * Dd, ba + l * Dd, hout);
        float* t = hin; hin = hout; hout = t;
    }

    // node aggregation + scores
    zero_f32<<<4096, 256, 0, stream>>>(nemb, (long)Bd * Nd * Dd);
    zero_f32<<<256, 256, 0, stream>>>(cnt, Nd);
    count_kernel<<<(NEVd + 255) / 256, 256, 0, stream>>>(cnt, evt);
    nodeagg_kernel<<<(int)((BNEVD + 255) / 256), 256, 0, stream>>>(nemb, hin, evt);
    score_kernel<<<(Bd * Nd + 255) / 256, 256, 0, stream>>>(nemb, cnt, Wo, bo, x, scr);

    // log-softmax
    logsoftmax_kernel<<<Bd, 1024, 0, stream>>>(scr, out);
}